// CenterNet_69956427317971
// MI455X (gfx1250) — compile-verified
//
#include <hip/hip_runtime.h>
#include <stdint.h>

// ---------------- problem constants ----------------
#define NUM_CLASSES 80
#define HDIM 256
#define WDIM 256
#define HW_ (HDIM * WDIM)
#define TOPK_ 100
#define STRIDE_ 4
#define INPUT_SIZE_ 1024

// tile geometry for the 5x5 stencil
#define TS 64
#define HALO 2
#define TW (TS + 2 * HALO) // 68
#define CPB 4              // classes per block (pipelined over z)

// ---------------- workspace layout (uint32 units) ----------------
#define WS_HIST 0        // 65536 bins
#define WS_COUNTER 65536 // candidate count
#define WS_GCNT 65537    // strictly-greater slot counter
#define WS_TCNT 65538    // tie slot counter
#define WS_TR 65539      // [T, R]
#define WS_SEL 65544     // 100 x uint2 (200 u32), 8B aligned
#define WS_ZERO_WORDS (WS_SEL + 200)
#define WS_CAND_BYTE 263168 // 16B aligned start of candidate buffer

typedef __attribute__((ext_vector_type(4))) unsigned int u32x4;
typedef __attribute__((ext_vector_type(4))) int i32x4;
typedef __attribute__((ext_vector_type(8))) int i32x8;

__global__ void cn_init_kernel(uint32_t* ws) {
  unsigned i = blockIdx.x * blockDim.x + threadIdx.x;
  if (i < WS_ZERO_WORDS) ws[i] = 0u;
}

// Issue one TDM 2D tile load: 68x68 f32 tile of a 256x256 plane -> LDS.
// D# built per CDNA5 ISA 8.3/8.4; OOB rows/cols on the right/bottom read 0
// (harmless: never consumed, windows are clamped to the image).
__device__ __forceinline__ void tdm_issue_tile(const float* src_origin,
                                               void* lds_dst) {
  unsigned long long ga = (unsigned long long)(const void*)src_origin;
  unsigned lds_addr = (unsigned)(unsigned long long)lds_dst;

  u32x4 g0;
  g0.x = 1u;       // count=1, user mode, no gather
  g0.y = lds_addr; // bits 63:32 lds byte address
  g0.z = (unsigned)ga;
  g0.w = (unsigned)((ga >> 32) & 0x1FFFFFFu) | 0x80000000u; // addr[56:32]|type=2

  i32x8 g1;
  g1[0] = (int)(2u << 16);             // data_size=2 (4B), wg_mask=0
  g1[1] = (int)((unsigned)WDIM << 16); // tensor_dim0[15:0]
  g1[2] = (int)((unsigned)HDIM << 16); // tensor_dim1[15:0]
  g1[3] = (int)((unsigned)TW << 16);   // tile_dim0 = 68
  g1[4] = (int)TW;                     // tile_dim1 = 68, tile_dim2 = 0
  g1[5] = (int)WDIM;                   // tensor_dim0_stride = 256 elements
  g1[6] = 0;
  g1[7] = 0;

  i32x4 gz = (i32x4){0, 0, 0, 0};
#if defined(__clang_major__) && (__clang_major__ >= 23)
  i32x8 gz8 = (i32x8){0, 0, 0, 0, 0, 0, 0, 0};
  __builtin_amdgcn_tensor_load_to_lds(g0, g1, gz, gz, gz8, 0);
#else
  __builtin_amdgcn_tensor_load_to_lds(g0, g1, gz, gz, 0);
#endif
}

// Fused, TDM-double-buffered: per (x,y) tile iterate CPB classes, issuing the
// next class's tile DMA while the current one is consumed (TENSORcnt is
// in-order per wave, so s_wait_tensorcnt(1) == "oldest load landed").
__global__ void cn_peaks_kernel(const float* __restrict__ cls,
                                uint32_t* __restrict__ hist,
                                uint32_t* __restrict__ counter,
                                uint2* __restrict__ cand, uint32_t cap) {
  __shared__ float tile[2][TW * TW];

  const int cbase = blockIdx.z * CPB;
  const int tx0 = blockIdx.x * TS;
  const int ty0 = blockIdx.y * TS;
  // clamp tile origin at left/top edges; windows are clamped per-pixel below
  const int sx = (tx0 == 0) ? 0 : tx0 - HALO;
  const int sy = (ty0 == 0) ? 0 : ty0 - HALO;
  const size_t tile_off = (size_t)sy * WDIM + sx;

  if (threadIdx.x < 32) { // wave 0 runs the TDM pipeline
    tdm_issue_tile(cls + (size_t)cbase * HW_ + tile_off, &tile[0][0]);
  }

  for (int k = 0; k < CPB; ++k) {
    const int c = cbase + k;
    if (threadIdx.x < 32) {
      if (k + 1 < CPB) { // prefetch next class into the other buffer
        tdm_issue_tile(cls + (size_t)(c + 1) * HW_ + tile_off,
                       &tile[(k + 1) & 1][0]);
        __builtin_amdgcn_s_wait_tensorcnt(1); // buffer k landed
      } else {
        __builtin_amdgcn_s_wait_tensorcnt(0);
      }
    }
    __syncthreads();

    const float* buf = &tile[k & 1][0];

    // each thread handles 16 interior pixels
    for (int i = threadIdx.x; i < TS * TS; i += blockDim.x) {
      const int px = i & (TS - 1);
      const int py = i >> 6;
      const int gx = tx0 + px;
      const int gy = ty0 + py;
      const float v = buf[(gy - sy) * TW + (gx - sx)];

      // replicate-clamped taps: duplicates are idempotent under max, so this
      // equals the SAME-padded (-inf) window max, but fully unrolls.
      int lx[5], lyr[5];
#pragma unroll
      for (int d = 0; d < 5; ++d) {
        int xx = gx + d - 2;
        xx = (xx < 0) ? 0 : ((xx > WDIM - 1) ? WDIM - 1 : xx);
        lx[d] = xx - sx;
        int yy = gy + d - 2;
        yy = (yy < 0) ? 0 : ((yy > HDIM - 1) ? HDIM - 1 : yy);
        lyr[d] = (yy - sy) * TW;
      }
      float m = -1e30f;
#pragma unroll
      for (int dy = 0; dy < 5; ++dy) {
        const int row = lyr[dy];
#pragma unroll
        for (int dx = 0; dx < 5; ++dx) m = fmaxf(m, buf[row + lx[dx]]);
      }

      if (v >= m) { // v is in the window, so v >= m <=> v == m (peak kept)
        const float s = 1.0f / (1.0f + __expf(-v));
        const unsigned key = __float_as_uint(s); // s in (0,1): monotone bits
        const unsigned pos = atomicAdd(counter, 1u);
        if (pos < cap) {
          cand[pos] = make_uint2(key, (unsigned)((c << 16) | (gy << 8) | gx));
          atomicAdd(&hist[key >> 16], 1u);
        }
      }
    }
    __syncthreads(); // buffer k&1 free for reuse at iteration k+1's prefetch
  }
}

// Suffix-scan the 65536-bin histogram from the top to find the threshold
// bin T and the number R of entries to take from bin T.
__global__ void cn_scan_kernel(const uint32_t* __restrict__ hist,
                               uint32_t* __restrict__ tr) {
  if (threadIdx.x == 0 && blockIdx.x == 0) {
    unsigned acc = 0, T = 0, R = 0;
    int done = 0;
    for (int b = 65535; b >= 0; --b) {
      const unsigned cnt = hist[b];
      if (acc + cnt >= TOPK_) {
        T = (unsigned)b;
        R = TOPK_ - acc;
        done = 1;
        break;
      }
      acc += cnt;
    }
    if (!done) { T = 0; R = TOPK_ - acc; } // degenerate: <100 peaks
    tr[0] = T;
    tr[1] = R;
  }
}

// Compact exactly TOPK_ winners: bins > T fill slots [0, 100-R) from the
// front, bin == T ties fill [100-R, 100) from the back.
__global__ void cn_compact_kernel(const uint2* __restrict__ cand,
                                  const uint32_t* __restrict__ counter,
                                  const uint32_t* __restrict__ tr,
                                  uint2* __restrict__ sel,
                                  uint32_t* __restrict__ gcnt,
                                  uint32_t* __restrict__ tcnt, uint32_t cap) {
  const unsigned total = *counter;
  const unsigned n = (total < cap) ? total : cap;
  const unsigned T = tr[0];
  const unsigned R = tr[1];
  for (unsigned i = blockIdx.x * blockDim.x + threadIdx.x; i < n;
       i += gridDim.x * blockDim.x) {
    const uint2 e = cand[i];
    const unsigned bin = e.x >> 16;
    if (bin > T) {
      const unsigned p = atomicAdd(gcnt, 1u);
      if (p < TOPK_) sel[p] = e;
    } else if (bin == T) {
      const unsigned p = atomicAdd(tcnt, 1u);
      if (p < R) sel[(TOPK_ - 1u) - p] = e;
    }
  }
}

// Rank-sort the 100 winners (O(100^2) compares, trivial), decode bboxes for
// batch 0, and write [bbox(100x4) | scores(100) | classes(100)] as f32.
__global__ void cn_final_kernel(const uint2* __restrict__ sel,
                                const float* __restrict__ txty,
                                float* __restrict__ out) {
  __shared__ unsigned skey[TOPK_];
  __shared__ unsigned sidx[TOPK_];
  const int t = threadIdx.x;
  if (t < TOPK_) {
    skey[t] = sel[t].x;
    sidx[t] = sel[t].y;
  }
  __syncthreads();
  if (t < TOPK_) {
    const unsigned k = skey[t];
    const unsigned id = sidx[t];
    int rank = 0;
    for (int j = 0; j < TOPK_; ++j) {
      const unsigned kj = skey[j];
      if (kj > k || (kj == k && j < t)) ++rank;
    }
    const float score = __uint_as_float(k);
    const unsigned cls_id = id >> 16;
    const unsigned pix = id & 0xFFFFu;
    const float gx = (float)(pix & 255u);
    const float gy = (float)(pix >> 8);
    const float tx = txty[pix];       // batch 0, channel 0
    const float ty = txty[HW_ + pix]; // batch 0, channel 1
    const float sx = 1.0f / (1.0f + __expf(-tx));
    const float sy = 1.0f / (1.0f + __expf(-ty));
    float bx = (sx + gx) * (float)STRIDE_ * (1.0f / (float)INPUT_SIZE_);
    float by = (sy + gy) * (float)STRIDE_ * (1.0f / (float)INPUT_SIZE_);
    bx = fminf(fmaxf(bx, 0.0f), 1.0f);
    by = fminf(fmaxf(by, 0.0f), 1.0f);
    out[rank * 4 + 0] = bx;
    out[rank * 4 + 1] = by;
    out[rank * 4 + 2] = 0.0f;
    out[rank * 4 + 3] = 0.0f;
    out[4 * TOPK_ + rank] = score;
    out[5 * TOPK_ + rank] = (float)cls_id;
  }
}

extern "C" void kernel_launch(void* const* d_in, const int* in_sizes, int n_in,
                              void* d_out, int out_size, void* d_ws,
                              size_t ws_size, hipStream_t stream) {
  const float* cls = (const float*)d_in[0];  // (8,80,256,256) f32; batch0 first
  const float* txty = (const float*)d_in[1]; // (8,2,256,256) f32
  float* out = (float*)d_out;                // 600 f32

  uint32_t* ws = (uint32_t*)d_ws;
  uint32_t* hist = ws + WS_HIST;
  uint32_t* counter = ws + WS_COUNTER;
  uint32_t* gcnt = ws + WS_GCNT;
  uint32_t* tcnt = ws + WS_TCNT;
  uint32_t* tr = ws + WS_TR;
  uint2* sel = (uint2*)(ws + WS_SEL);
  uint2* cand = (uint2*)((char*)d_ws + WS_CAND_BYTE);

  uint32_t cap = 0;
  if (ws_size > (size_t)WS_CAND_BYTE + 8) {
    size_t c = (ws_size - WS_CAND_BYTE) / sizeof(uint2);
    cap = (uint32_t)((c > (size_t)(1u << 20)) ? (1u << 20) : c);
  }

  cn_init_kernel<<<(WS_ZERO_WORDS + 255) / 256, 256, 0, stream>>>(ws);

  dim3 pg(WDIM / TS, HDIM / TS, NUM_CLASSES / CPB); // 4 x 4 x 20
  cn_peaks_kernel<<<pg, 256, 0, stream>>>(cls, hist, counter, cand, cap);

  cn_scan_kernel<<<1, 32, 0, stream>>>(hist, tr);
  cn_compact_kernel<<<256, 256, 0, stream>>>(cand, counter, tr, sel, gcnt,
                                             tcnt, cap);
  cn_final_kernel<<<1, 128, 0, stream>>>(sel, txty, out);
}